// DifferentiableVACF_74105365725348
// MI455X (gfx1250) — compile-verified
//
#include <hip/hip_runtime.h>

// ---------------------------------------------------------------------------
// VACF = band of Gram matrix G = F F^T (F = vel.reshape(T, D)), lags 0..W-1.
// 16x16 fp32 WMMA tiles on diagonals d = 0..7; double-buffered async
// Global->LDS staging (ASYNCcnt) overlapped with the WMMA K-loop.
// ---------------------------------------------------------------------------

#define T_DIM   10000
#define D_DIM   3000
#define KC      40                 // K per chunk (10 float4 per row)
#define STRIDE  44                 // padded LDS row stride: conflict-free banks
#define ROWS    128                // 8 row-tiles of 16 staged per workgroup
#define KLEN    600                // K per grid.y split = 15 full chunks
#define KSPLIT  5                  // 5 * 600 = 3000, no partial chunks
#define NCH     (KLEN / KC)        // 15
#define F4_PER_THREAD (ROWS * (KC / 4) / 256)   // 5 async b128 per thread/chunk

typedef __attribute__((ext_vector_type(2))) float v2f;
typedef __attribute__((ext_vector_type(8))) float v8f;
typedef __attribute__((ext_vector_type(4))) int   v4i;

#define AS_GLOBAL __attribute__((address_space(1)))
#define AS_SHARED __attribute__((address_space(3)))
typedef AS_GLOBAL v4i* gv4i_p;     // global int4*
typedef AS_SHARED v4i* sv4i_p;     // LDS int4*

#if defined(__has_builtin)
#if __has_builtin(__builtin_amdgcn_global_load_async_to_lds_b128)
#define HAVE_ASYNC_LDS 1
#endif
#endif
#ifndef HAVE_ASYNC_LDS
#define HAVE_ASYNC_LDS 0
#endif

#if HAVE_ASYNC_LDS
#if __has_builtin(__builtin_amdgcn_s_wait_asynccnt)
#define WAIT_ASYNC(n) __builtin_amdgcn_s_wait_asynccnt(n)
#else
#define WAIT_ASYNC(n) asm volatile("s_wait_asynccnt %0" ::"n"(n) : "memory")
#endif
#else
#define WAIT_ASYNC(n)
#endif

__global__ void __launch_bounds__(256)
vacf_zero_ws(float* __restrict__ p, int n) {
    int i = blockIdx.x * blockDim.x + threadIdx.x;
    if (i < n) p[i] = 0.0f;
}

__global__ void __launch_bounds__(256)
vacf_band_wmma(const float* __restrict__ flat,   // [T_DIM, D_DIM] fp32
               float* __restrict__ lagAcc,       // [>=128] fp32 accumulators
               int W) {                          // vacf_window (== out_size)
    __shared__ __align__(16) float lds[2][ROWS * STRIDE];  // double buffer
    __shared__ float ldsLag[ROWS];

    const int I    = blockIdx.x;       // row-tile index (0..624)
    const int tid  = threadIdx.x;
    const int wave = tid >> 5;         // wave32: wave id == diagonal offset d
    const int lane = tid & 31;
    const int m    = lane & 15;        // M (A) / N (B) index for this lane
    const int kh   = lane >> 4;        // K half-select (K 0,1 vs 2,3)

    if (tid < ROWS) ldsLag[tid] = 0.0f;

    const int d = wave;                // tile-diagonal offset
    const int J = I - d;               // column tile
    const bool active = (J >= 0) && (16 * d - 15 <= W - 1);

    const int rowBase = 16 * I - (ROWS - 16);   // global row of LDS row 0
    const int kbeg    = blockIdx.y * KLEN;

    const int rA = ((ROWS - 16) + m) * STRIDE;          // A rows: tile I
    const int rB = ((ROWS - 16) - 16 * d + m) * STRIDE; // B rows: tile J

    // issue one chunk's staging: 5 x b128 per thread, uniform count
    auto issue = [&](int chunk, float* buf) {
        const int k0 = kbeg + chunk * KC;
        #pragma unroll
        for (int it = 0; it < F4_PER_THREAD; ++it) {
            const int q  = tid + it * 256;
            const int r  = q / (KC / 4);            // chunk row 0..127
            const int c4 = (q % (KC / 4)) << 2;     // element offset in row
            int g = rowBase + r;                    // clamp keeps counts uniform;
            g = g < 0 ? 0 : g;                      // rows g<0 never read by active waves
            const float* src = flat + (size_t)g * D_DIM + (k0 + c4);
            float* dst = buf + r * STRIDE + c4;
#if HAVE_ASYNC_LDS
            __builtin_amdgcn_global_load_async_to_lds_b128(
                (gv4i_p)src, (sv4i_p)dst, 0, 0);
#else
            *(float4*)dst = *(const float4*)src;
#endif
        }
    };

    v8f c = {0.f, 0.f, 0.f, 0.f, 0.f, 0.f, 0.f, 0.f};

    issue(0, lds[0]);                               // prologue
    for (int i = 0; i < NCH; ++i) {
        if (i + 1 < NCH) {
            issue(i + 1, lds[(i + 1) & 1]);         // overlap next chunk's DMA
            WAIT_ASYNC(F4_PER_THREAD);              // chunk i complete, i+1 in flight
        } else {
            WAIT_ASYNC(0);
        }
        __syncthreads();                            // chunk i visible to all waves

        if (active) {                               // wave-uniform: EXEC all 1s
            const float* buf = lds[i & 1];
            #pragma unroll
            for (int ks = 0; ks < KC; ks += 4) {
                // A 16x4: lane m holds K = 2*kh, 2*kh+1; B 4x16 mirrors it
                v2f a = *(const v2f*)(buf + rA + ks + 2 * kh);
                v2f b = *(const v2f*)(buf + rB + ks + 2 * kh);
                c = __builtin_amdgcn_wmma_f32_16x16x4_f32(
                        false, a, false, b, (short)0, c, false, false);
            }
        }
        __syncthreads();                            // buf reusable at i+2
    }

    // scatter C tile into per-lag sums: C[r] = G[16I + M, 16J + N]
    if (active) {
        #pragma unroll
        for (int r = 0; r < 8; ++r) {
            const int M   = r + 8 * kh;      // lanes 16-31 hold M+8
            const int lag = 16 * d + M - m;  // N == m
            if (lag >= 0 && lag < W) {
                atomicAdd(&ldsLag[lag], c[r]);       // ds_add_f32
            }
        }
    }
    __syncthreads();
    if (tid < W) {
        atomicAdd(&lagAcc[tid], ldsLag[tid]);        // global_atomic_add_f32
    }
}

__global__ void __launch_bounds__(256)
vacf_finalize(const float* __restrict__ acc, float* __restrict__ out, int W) {
    int i = blockIdx.x * blockDim.x + threadIdx.x;
    if (i < W) {
        double cnt = (double)(T_DIM - i) * (double)D_DIM;
        out[i] = (float)((double)acc[i] / cnt);
    }
}

extern "C" void kernel_launch(void* const* d_in, const int* in_sizes, int n_in,
                              void* d_out, int out_size, void* d_ws, size_t ws_size,
                              hipStream_t stream) {
    (void)in_sizes; (void)n_in; (void)ws_size;
    const float* vel = (const float*)d_in[0];    // [10000, 1000, 3] fp32
    const int W = out_size;                      // = vacf_window = 100
    float* acc = (float*)d_ws;                   // 128 fp32 lag accumulators

    vacf_zero_ws<<<1, 256, 0, stream>>>(acc, ROWS);

    dim3 grid(T_DIM / 16, KSPLIT);               // 625 x 5 workgroups
    vacf_band_wmma<<<grid, 256, 0, stream>>>(vel, acc, W);

    vacf_finalize<<<(W + 255) / 256, 256, 0, stream>>>(acc, (float*)d_out, W);
}